// ICP_o3d_1400159339067
// MI455X (gfx1250) — compile-verified
//
#include <hip/hip_runtime.h>
#include <stdint.h>

typedef __attribute__((ext_vector_type(2))) float v2f;
typedef __attribute__((ext_vector_type(8))) float v8f;

#define B_   8
#define M_   2048
#define N_   2048
#define THR2 0.25f

// Workspace float layout
#define WS_R    0      // 72 floats: per-batch 3x3 R
#define WS_T    72     // 24 floats: per-batch t
#define WS_ACC  96     // 18 floats per batch: Sw, Sp[3], Sq[3], Spq[9], Sd2, pad
#define WS_ANY  240    // 8 floats: any_valid flag per batch
#define WS_DEST 256    // B*N*4 floats: augmented dest points

__global__ void zero_kernel(const float* __restrict__ initR,
                            const float* __restrict__ initT,
                            float* __restrict__ ws) {
  int t = threadIdx.x;
  if (t < 72)  ws[WS_R + t] = initR[t];
  if (t < 24)  ws[WS_T + t] = initT[t];
  if (t < 152) ws[WS_ACC + t] = 0.0f;   // accumulators + any_valid region
}

__global__ void prep_kernel(const float* __restrict__ dest, float* __restrict__ ws) {
  int i = blockIdx.x * blockDim.x + threadIdx.x;
  if (i >= B_ * N_) return;
  int b = i / N_, n = i % N_;
  const float* d = dest + b * 3 * N_;
  float x = d[n], y = d[N_ + n], z = d[2 * N_ + n];
  bool valid = (x != 0.0f) && (y != 0.0f) && (z != 0.0f);
  float* da = ws + WS_DEST + (size_t)i * 4;
  da[0] = x; da[1] = y; da[2] = z;
  da[3] = valid ? (x * x + y * y + z * z) : __builtin_inff();
  if (valid) ws[WS_ANY + b] = 1.0f;     // racy identical writes: fine
}

// Correspondence + accumulation. mode 0: moment accumulation for Kabsch.
// mode 1: final pass, accumulate Sw and sum(w*d2) for RMSE.
__global__ __launch_bounds__(256) void corr_kernel(const float* __restrict__ src,
                                                   float* __restrict__ ws, int mode) {
  const int b    = blockIdx.y;
  const int wave = threadIdx.x >> 5;
  const int lane = threadIdx.x & 31;
  const int col  = lane & 15;          // N within tile / M-row within tile
  const int half = lane >> 4;
  const int m_base = blockIdx.x * 128 + wave * 16;

  // ---- Stage dest_aug[b] (32 KB) into LDS once per block via async copy ----
  __shared__ float lds_dest[N_ * 4];
  {
    const float* gsrc = ws + WS_DEST + (size_t)b * N_ * 4;
#pragma unroll
    for (int i = 0; i < 8; ++i) {
      int off = (threadIdx.x + i * 256) * 4;     // float index; 16B per thread
      uint64_t ga = (uint64_t)(uintptr_t)(gsrc + off);
      uint32_t la = (uint32_t)(uintptr_t)(&lds_dest[off]);
      asm volatile("global_load_async_to_lds_b128 %0, %1, off"
                   :: "v"(la), "v"(ga) : "memory");
    }
    asm volatile("s_wait_asynccnt 0" ::: "memory");
    __syncthreads();
  }

  float R[9], tv[3];
#pragma unroll
  for (int i = 0; i < 9; ++i) R[i] = ws[WS_R + b * 9 + i];
#pragma unroll
  for (int i = 0; i < 3; ++i) tv[i] = ws[WS_T + b * 3 + i];

  // A operand: row m = m_base + col; A row = [-2px, -2py, -2pz, 1]
  // ISA layout: lanes 0-15 supply K0,K1 in (v0,v1); lanes 16-31 supply K2,K3.
  int m = m_base + col;
  float sx = src[m], sy = src[M_ + m], sz = src[2 * M_ + m];
  float px = R[0] * sx + R[1] * sy + R[2] * sz + tv[0];
  float py = R[3] * sx + R[4] * sy + R[5] * sz + tv[1];
  float pz = R[6] * sx + R[7] * sy + R[8] * sz + tv[2];
  v2f a;
  a.x = half ? (-2.0f * pz) : (-2.0f * px);
  a.y = half ? 1.0f         : (-2.0f * py);

  float rmin[8];
  int   ridx[8];
#pragma unroll
  for (int j = 0; j < 8; ++j) { rmin[j] = __builtin_inff(); ridx[j] = 0x7fffffff; }

  // Per-lane B-operand source in LDS: element n = n0+col, halves read
  // (qx,qy) or (qz,|q|^2) -> 8-byte LDS loads with immediate offsets.
  const float* lp = &lds_dest[col * 4 + half * 2];

#pragma unroll 4
  for (int n0 = 0; n0 < N_; n0 += 16) {
    v2f bb = *(const v2f*)(lp + n0 * 4);
    v8f c = {};
    c = __builtin_amdgcn_wmma_f32_16x16x4_f32(false, a, false, bb, (short)0, c,
                                              false, false);
    const int n = n0 + col;
#pragma unroll
    for (int j = 0; j < 8; ++j) {
      float s = c[j];   // score for row (m_base + half*8 + j), column n
      if (s < rmin[j] || (s == rmin[j] && n < ridx[j])) { rmin[j] = s; ridx[j] = n; }
    }
  }

  // argmin reduction across the 16 lanes of each half-group
#pragma unroll
  for (int j = 0; j < 8; ++j) {
    for (int off = 8; off; off >>= 1) {
      float om = __shfl_xor(rmin[j], off, 16);
      int   oi = __shfl_xor(ridx[j], off, 16);
      if (om < rmin[j] || (om == rmin[j] && oi < ridx[j])) { rmin[j] = om; ridx[j] = oi; }
    }
  }

  float anyv = ws[WS_ANY + b];
  if (col == 0) {   // lanes 0 and 16 finalize 8 rows each
    float acc[16];
#pragma unroll
    for (int i = 0; i < 16; ++i) acc[i] = 0.0f;
    float accSw = 0.0f, accSd2 = 0.0f;

    for (int j = 0; j < 8; ++j) {
      int mr = m_base + half * 8 + j;
      float ssx = src[mr], ssy = src[M_ + mr], ssz = src[2 * M_ + mr];
      float ppx = R[0] * ssx + R[1] * ssy + R[2] * ssz + tv[0];
      float ppy = R[3] * ssx + R[4] * ssy + R[5] * ssz + tv[1];
      float ppz = R[6] * ssx + R[7] * ssy + R[8] * ssz + tv[2];
      const float* q = &lds_dest[ridx[j] * 4];
      float qx = q[0], qy = q[1], qz = q[2];
      float dx = ppx - qx, dy = ppy - qy, dz = ppz - qz;
      float d2 = fmaxf(dx * dx + dy * dy + dz * dz, 0.0f);
      float w = ((d2 < THR2) && (anyv > 0.0f)) ? 1.0f : 0.0f;
      if (mode == 0) {
        acc[0]  += w;
        acc[1]  += w * ssx; acc[2]  += w * ssy; acc[3]  += w * ssz;
        acc[4]  += w * qx;  acc[5]  += w * qy;  acc[6]  += w * qz;
        acc[7]  += w * ssx * qx; acc[8]  += w * ssx * qy; acc[9]  += w * ssx * qz;
        acc[10] += w * ssy * qx; acc[11] += w * ssy * qy; acc[12] += w * ssy * qz;
        acc[13] += w * ssz * qx; acc[14] += w * ssz * qy; acc[15] += w * ssz * qz;
      } else {
        accSw += w; accSd2 += w * d2;
      }
    }
    float* A = ws + WS_ACC + b * 18;
    if (mode == 0) {
      for (int i = 0; i < 16; ++i) atomicAdd(&A[i], acc[i]);
    } else {
      atomicAdd(&A[0], accSw);
      atomicAdd(&A[16], accSd2);
    }
  }
}

// Per-batch Kabsch via Horn's quaternion (power iteration on shifted 4x4).
__global__ void solve_kernel(float* __restrict__ ws) {
  int b = threadIdx.x;
  if (b >= B_) return;
  float* A = ws + WS_ACC + b * 18;
  float Sw = A[0];
  float Sp[3] = { A[1], A[2], A[3] };
  float Sq[3] = { A[4], A[5], A[6] };
  float Spq[9];
  for (int i = 0; i < 9; ++i) Spq[i] = A[7 + i];
  for (int i = 0; i < 18; ++i) A[i] = 0.0f;   // reset for next pass
  if (!(Sw > 0.0f)) return;                   // keep previous R, t

  float safe = fmaxf(Sw, 1.0f);
  float mp[3], mq[3];
  for (int i = 0; i < 3; ++i) { mp[i] = Sp[i] / safe; mq[i] = Sq[i] / safe; }
  float H[9];
  for (int i = 0; i < 3; ++i)
    for (int j = 0; j < 3; ++j)
      H[i * 3 + j] = Spq[i * 3 + j] - Sp[i] * Sq[j] / safe;

  float Sxx = H[0], Sxy = H[1], Sxz = H[2];
  float Syx = H[3], Syy = H[4], Syz = H[5];
  float Szx = H[6], Szy = H[7], Szz = H[8];
  float Nm[16] = {
    Sxx + Syy + Szz, Syz - Szy,        Szx - Sxz,        Sxy - Syx,
    Syz - Szy,       Sxx - Syy - Szz,  Sxy + Syx,        Szx + Sxz,
    Szx - Sxz,       Sxy + Syx,       -Sxx + Syy - Szz,  Syz + Szy,
    Sxy - Syx,       Szx + Sxz,        Syz + Szy,       -Sxx - Syy + Szz };
  float fn = 0.0f;
  for (int i = 0; i < 16; ++i) fn += Nm[i] * Nm[i];
  float shift = sqrtf(fn) + 1e-12f;
  for (int i = 0; i < 4; ++i) Nm[i * 4 + i] += shift;  // make top eig dominant

  float v[4] = { 1.0f, 0.0123f, 0.0456f, 0.0789f };
  for (int it = 0; it < 60; ++it) {
    float nv[4];
    for (int i = 0; i < 4; ++i)
      nv[i] = Nm[i*4+0]*v[0] + Nm[i*4+1]*v[1] + Nm[i*4+2]*v[2] + Nm[i*4+3]*v[3];
    float nrm = sqrtf(nv[0]*nv[0] + nv[1]*nv[1] + nv[2]*nv[2] + nv[3]*nv[3]) + 1e-30f;
    for (int i = 0; i < 4; ++i) v[i] = nv[i] / nrm;
  }
  float qw = v[0], qx = v[1], qy = v[2], qz = v[3];
  float Rn[9] = {
    1 - 2*(qy*qy + qz*qz), 2*(qx*qy - qw*qz),     2*(qx*qz + qw*qy),
    2*(qx*qy + qw*qz),     1 - 2*(qx*qx + qz*qz), 2*(qy*qz - qw*qx),
    2*(qx*qz - qw*qy),     2*(qy*qz + qw*qx),     1 - 2*(qx*qx + qy*qy) };
  float tn[3];
  for (int i = 0; i < 3; ++i)
    tn[i] = mq[i] - (Rn[i*3] * mp[0] + Rn[i*3+1] * mp[1] + Rn[i*3+2] * mp[2]);
  for (int i = 0; i < 9; ++i) ws[WS_R + b * 9 + i] = Rn[i];
  for (int i = 0; i < 3; ++i) ws[WS_T + b * 3 + i] = tn[i];
}

__global__ void final_kernel(const float* __restrict__ ws, float* __restrict__ out) {
  int t = threadIdx.x;
  if (t < 72) out[t]      = ws[WS_R + t];
  if (t < 24) out[72 + t] = ws[WS_T + t];
  if (t < 8) {
    const float* A = ws + WS_ACC + t * 18;
    float Sw = A[0], Sd2 = A[16];
    out[96 + t] = (Sw > 0.0f) ? sqrtf(Sd2 / fmaxf(Sw, 1.0f))
                              : __builtin_nanf("");
  }
}

extern "C" void kernel_launch(void* const* d_in, const int* in_sizes, int n_in,
                              void* d_out, int out_size, void* d_ws, size_t ws_size,
                              hipStream_t stream) {
  const float* src  = (const float*)d_in[0];   // (B,3,M) -- batch 0 used
  const float* dest = (const float*)d_in[1];   // (B,3,N)
  const float* iR   = (const float*)d_in[2];   // (B,3,3)
  const float* iT   = (const float*)d_in[3];   // (B,3,1)
  float* ws  = (float*)d_ws;
  float* out = (float*)d_out;

  zero_kernel<<<1, 256, 0, stream>>>(iR, iT, ws);
  prep_kernel<<<(B_ * N_ + 255) / 256, 256, 0, stream>>>(dest, ws);

  dim3 grid(M_ / 128, B_);
  for (int it = 0; it < 10; ++it) {
    corr_kernel<<<grid, 256, 0, stream>>>(src, ws, 0);
    solve_kernel<<<1, 32, 0, stream>>>(ws);
  }
  corr_kernel<<<grid, 256, 0, stream>>>(src, ws, 1);  // final: Sw, sum(w*d2)
  final_kernel<<<1, 128, 0, stream>>>(ws, out);
}